// ModelNew_32753420600120
// MI455X (gfx1250) — compile-verified
//
#include <hip/hip_runtime.h>
#include <stdint.h>

typedef __attribute__((ext_vector_type(2))) float v2f;
typedef __attribute__((ext_vector_type(4))) float v4f;
typedef __attribute__((ext_vector_type(8))) float v8f;

#define NN 4096
#define BT 128        // C tile per workgroup (M and N)
#define KT 32         // K chunk
#define A_STR 36      // KT + 4 pad  (16B-aligned rows, conflict-free frag reads)
#define BQ_STR 288    // floats per B k-pair group: 2*128 interleaved + 32 pad

// CDNA5 async global->LDS copy (ASYNCcnt path). GVS mode:
//   copies 16 bytes from the SGPR base plus the VGPR byte offset to the LDS address in a VGPR
__device__ __forceinline__ void async_ld_b128(uint32_t lds_addr, uint32_t goff,
                                              const float* base) {
  asm volatile("global_load_async_to_lds_b128 %0, %1, %2"
               :
               : "v"(lds_addr), "v"(goff), "s"(base)
               : "memory");
}
__device__ __forceinline__ void wait_async0() {
  asm volatile("s_wait_asynccnt 0x0" ::: "memory");
}

__global__ __launch_bounds__(256) void tril_gemm_wmma_f32(
    const float* __restrict__ A, const float* __restrict__ B,
    float* __restrict__ C) {
  const int bi = blockIdx.y;     // row block
  const int bj = blockIdx.x;     // col block
  const int tid = threadIdx.x;
  const int row0 = bi * BT;
  const int col0 = bj * BT;

  if (bj > bi) {
    // strictly-upper tile of tril(): write zeros (d_out is poisoned)
    const v4f z = {0.f, 0.f, 0.f, 0.f};
    for (int f = tid; f < BT * BT / 4; f += 256) {
      int r = f >> 5;
      int c = (f & 31) << 2;
      *(v4f*)&C[(size_t)(row0 + r) * NN + col0 + c] = z;
    }
    return;
  }

  __shared__ float Alds[2][BT * A_STR];       // 2 x 18KB, row-major + pad
  __shared__ float Blds[2][(KT / 2) * BQ_STR];// 2 x 18KB, K-pair interleaved

  const int wave  = tid >> 5;             // 0..7
  const int lane  = tid & 31;
  const int lr    = lane & 15;            // M (A/C) or N (B) index within 16
  const int khalf = (lane >> 4) << 1;     // 0 or 2 : K pair this half-wave owns
  const int wm = (wave >> 2) * 64;        // wave M offset: 0 / 64
  const int wn = (wave & 3)  * 32;        // wave N offset: 0/32/64/96

  // Raw LDS byte offsets (flat LDS address carries offset in addr[31:0])
  const uint32_t ldsA[2] = {(uint32_t)(uintptr_t)&Alds[0][0],
                            (uint32_t)(uintptr_t)&Alds[1][0]};
  const uint32_t ldsB[2] = {(uint32_t)(uintptr_t)&Blds[0][0],
                            (uint32_t)(uintptr_t)&Blds[1][0]};

  v8f acc[4][2];
  const v8f vz = {0.f,0.f,0.f,0.f,0.f,0.f,0.f,0.f};
#pragma unroll
  for (int mt = 0; mt < 4; ++mt)
#pragma unroll
    for (int nt = 0; nt < 2; ++nt) acc[mt][nt] = vz;

  // Triangular K range: outside it, A-tile or B-tile is identically zero.
  const int kmin = col0;                                   // bj*128
  const int kmax = (row0 + BT < NN) ? (row0 + BT) : NN;    // (bi+1)*128
  const int nChunks = (kmax - kmin) / KT;                  // >= 4, all full

  auto issue_async = [&](int kc, int buf) {
    const int k0 = kmin + kc * KT;
#pragma unroll
    for (int p = 0; p < 4; ++p) {
      const int f = tid + (p << 8);
      // A tile: 128 rows x 8 float4, row-major stride A_STR
      const int ar = f >> 3, ac = (f & 7) << 2;
      async_ld_b128(ldsA[buf] + (uint32_t)(ar * A_STR + ac) * 4u,
                    (uint32_t)(((row0 + ar) * NN + k0 + ac) * 4), A);
      // B tile: 32 rows x 32 float4 -> K-pair interleaved 16B chunks:
      //   LDS float offset = (k>>1)*BQ_STR + 2*n + (k&1)*4   (n mult of 4)
      const int br = f >> 5, bc = (f & 31) << 2;
      const int bOff = (br >> 1) * BQ_STR + (bc << 1) + ((br & 1) << 2);
      async_ld_b128(ldsB[buf] + (uint32_t)bOff * 4u,
                    (uint32_t)(((k0 + br) * NN + col0 + bc) * 4), B);
    }
  };

  issue_async(0, 0);
  wait_async0();
  __syncthreads();

  // Loop-invariant B fragment bases: element (k, n) at
  //   (k>>1)*BQ_STR + ((n>>2)<<3) + ((k&1)<<2) + (n&3);  k = ks*4 + khalf (+0/+1)
  const int qb = (khalf >> 1) * BQ_STR;   // 0 or BQ_STR (per-lane constant)
  int bbase[2];
#pragma unroll
  for (int nt = 0; nt < 2; ++nt) {
    const int cb = wn + nt * 16 + lr;
    bbase[nt] = qb + ((cb >> 2) << 3) + (cb & 3);
  }

  for (int kc = 0; kc < nChunks; ++kc) {
    // Prefetch next chunk into the other buffer (its last readers finished
    // at the barrier ending iteration kc-1).
    if (kc + 1 < nChunks) issue_async(kc + 1, (kc + 1) & 1);

    const int buf = kc & 1;
#pragma unroll
    for (int ks = 0; ks < KT / 4; ++ks) {
      // A frags: lane L -> row lr, K = ks*4 + khalf + {0,1} (ISA 16x4 layout)
      v2f afrag[4];
#pragma unroll
      for (int mt = 0; mt < 4; ++mt)
        afrag[mt] = *(const v2f*)&Alds[buf][(wm + mt * 16 + lr) * A_STR +
                                            (ks << 2) + khalf];
      // B frags: .x = (k), .y = (k+1) -> 2 b32 loads 16B apart, adjacent VGPRs
      v2f bfrag[2];
#pragma unroll
      for (int nt = 0; nt < 2; ++nt) {
        const int o = bbase[nt] + ks * (2 * BQ_STR);
        bfrag[nt].x = Blds[buf][o];
        bfrag[nt].y = Blds[buf][o + 4];
      }
#pragma unroll
      for (int mt = 0; mt < 4; ++mt)
#pragma unroll
        for (int nt = 0; nt < 2; ++nt)
          acc[mt][nt] = __builtin_amdgcn_wmma_f32_16x16x4_f32(
              false, afrag[mt], false, bfrag[nt], (short)0, acc[mt][nt],
              false, false);
    }

    if (kc + 1 < nChunks) wait_async0();  // after compute: overlap DMA
    __syncthreads();
  }

  // Epilogue: C/D layout -> VGPR v, lane L : M = v + 8*(L>=16), N = L%16
  const bool diag = (bi == bj);
  const int mhi = (lane >> 4) << 3;
#pragma unroll
  for (int mt = 0; mt < 4; ++mt) {
#pragma unroll
    for (int nt = 0; nt < 2; ++nt) {
      const int j = col0 + wn + nt * 16 + lr;
#pragma unroll
      for (int v = 0; v < 8; ++v) {
        const int i = row0 + wm + mt * 16 + v + mhi;
        float val = acc[mt][nt][v];
        if (diag && j > i) val = 0.f;   // tril mask on diagonal blocks
        C[(size_t)i * NN + j] = val;
      }
    }
  }
}

extern "C" void kernel_launch(void* const* d_in, const int* in_sizes, int n_in,
                              void* d_out, int out_size, void* d_ws, size_t ws_size,
                              hipStream_t stream) {
  const float* A = (const float*)d_in[0];
  const float* B = (const float*)d_in[1];
  float* C = (float*)d_out;
  (void)in_sizes; (void)n_in; (void)out_size; (void)d_ws; (void)ws_size;
  dim3 grid(NN / BT, NN / BT);   // 32 x 32 blocks; upper blocks just zero-fill
  dim3 block(256);
  tril_gemm_wmma_f32<<<grid, block, 0, stream>>>(A, B, C);
}